// WaveRNN_68942815036070
// MI455X (gfx1250) — compile-verified
//
#include <hip/hip_runtime.h>

// ---------------------------------------------------------------------------
// Persistent-kernel 2D FDTD wave propagator for MI455X (gfx1250).
//  * 64 workgroups x 256 threads (8 wave32 each); each WG owns an 8-row strip
//    of the 512x512 grid, both time levels resident in LDS (~40 KB).
//  * Horizontal 1-D second difference via V_WMMA_F32_16X16X4_F32:
//    P = S * D,  D = tridiag(1,-2,1) 16x16 (exact in fp32), where S stacks two
//    8x16 column blocks into the 16x16 A operand. Vertical diffs + block-edge
//    corrections done with per-lane VALU (co-executes with the matrix pipe).
//  * Halo rows exchanged through L2 (double-buffered in d_ws) each step;
//    incoming halos pulled with GLOBAL_LOAD_ASYNC_TO_LDS_B128 (ASYNCcnt).
//  * Device-wide sense-reversal barrier (agent-scope acquire atomics give the
//    required cache invalidation) -> one kernel launch for all 1200 steps.
// ---------------------------------------------------------------------------

#define NZv   512
#define NXv   512
#define NTv   1200
#define NSRCv 4
#define NRECv 512
#define DTf   0.001f
#define DHf   10.0f

#define NBLK  64          // workgroups (strips)
#define ROWS  8           // grid rows per strip
#define NTHR  256         // 8 wave32 per WG
#define NXP   516         // LDS row pitch (floats): 16B-aligned rows, bank skew

typedef __attribute__((ext_vector_type(2))) float v2f;
typedef __attribute__((ext_vector_type(4))) float f4v;
typedef __attribute__((ext_vector_type(8))) float v8f;
typedef __attribute__((ext_vector_type(4))) int   v4i;

#define AS1 __attribute__((address_space(1)))
#define AS3 __attribute__((address_space(3)))

typedef AS1 v4i* g_v4i_p;   // global (AS1) pointer to int4
typedef AS3 v4i* l_v4i_p;   // LDS (AS3) pointer to int4

#if defined(__AMDGCN__) && __has_builtin(__builtin_amdgcn_global_load_async_to_lds_b128) && __has_builtin(__builtin_amdgcn_s_wait_asynccnt)
#define USE_ASYNC 1
#else
#define USE_ASYNC 0
#endif

__device__ __forceinline__ void grid_barrier(unsigned* cnt, unsigned* gen) {
    __syncthreads();
    if (threadIdx.x == 0) {
        unsigned g = __hip_atomic_load(gen, __ATOMIC_RELAXED, __HIP_MEMORY_SCOPE_AGENT);
        if (__hip_atomic_fetch_add(cnt, 1u, __ATOMIC_ACQ_REL, __HIP_MEMORY_SCOPE_AGENT) == NBLK - 1u) {
            __hip_atomic_store(cnt, 0u, __ATOMIC_RELAXED, __HIP_MEMORY_SCOPE_AGENT);
            __hip_atomic_fetch_add(gen, 1u, __ATOMIC_ACQ_REL, __HIP_MEMORY_SCOPE_AGENT);
        } else {
            while (__hip_atomic_load(gen, __ATOMIC_ACQUIRE, __HIP_MEMORY_SCOPE_AGENT) == g)
                __builtin_amdgcn_s_sleep(1);
        }
    }
    __syncthreads();
}

__global__ void fdtd_init_ws(unsigned* bar) {
    if (threadIdx.x == 0) { bar[0] = 0u; bar[1] = 0u; }
}

__global__ __launch_bounds__(NTHR) void fdtd_wave_persist(
    const float* __restrict__ x,      // (NSRC, NT)
    const float* __restrict__ vel,    // (NZ, NX)
    const int*   __restrict__ src_z, const int* __restrict__ src_x,
    const int*   __restrict__ rec_z, const int* __restrict__ rec_x,
    float*       __restrict__ out,    // (NT, NREC)
    unsigned*    __restrict__ bar,    // [cnt, gen]
    float*       __restrict__ ghalo)  // 2 x NBLK x 2 x NX floats
{
    __shared__ float lds[2 * (ROWS + 2) * NXP];   // two time levels + halo rows

    const int tid  = threadIdx.x;
    const int wg   = blockIdx.x;
    const int z0   = wg * ROWS;
    const int lane = tid & 31;
    const int wave = tid >> 5;
    const int hf   = lane >> 4;      // half-wave: selects K/M upper pair
    const int li   = lane & 15;

    float* bufA = &lds[0];
    float* bufB = &lds[(ROWS + 2) * NXP];

    for (int i = tid; i < 2 * (ROWS + 2) * NXP; i += NTHR) lds[i] = 0.0f;

    // B operand = D (tridiag 1,-2,1), constant: 4 K-chunks of 4, per ISA layout
    // lanes 0-15 hold K = 4k+{0,1}, lanes 16-31 hold K = 4k+{2,3}, N = lane%16.
    v2f Bc[4];
#pragma unroll
    for (int k = 0; k < 4; ++k) {
        const int r0 = 4 * k + (hf ? 2 : 0);
        const int r1 = r0 + 1;
        Bc[k].x = (r0 == li) ? -2.f : ((r0 - li == 1 || li - r0 == 1) ? 1.f : 0.f);
        Bc[k].y = (r1 == li) ? -2.f : ((r1 - li == 1 || li - r1 == 1) ? 1.f : 0.f);
    }

    // Per-lane (vel*DT/DH)^2 for the 16 cells this lane owns (2 tiles x 8 rows)
    float cf_r[2][8];
#pragma unroll
    for (int i = 0; i < 2; ++i) {
        const int st = wave * 2 + i;                 // stacked tile index 0..15
        const int j  = st * 32 + hf * 16 + li;       // this lane's column
#pragma unroll
        for (int v = 0; v < 8; ++v) {
            const float c = vel[(z0 + v) * NXv + j] * (DTf / DHf);
            cf_r[i][v] = c * c;
        }
    }
    __syncthreads();

    float* h1 = bufA;                 // current field  (interior rows 1..8, halo 0/9)
    float* h2 = bufB;                 // previous field; overwritten with h_new

    const int side = tid >> 7;        // 0: top boundary row, 1: bottom
    const int seg  = tid & 127;       // 128 x float4 segments per 512-col row

    unsigned* cnt = bar;
    unsigned* gen = bar + 1;

#pragma unroll 1
    for (int t = 0; t < NTv; ++t) {
        // ---- h2 <- 2*h1 - h2 + cf * (dh^2 * laplacian(h1)) -------------------
#pragma unroll
        for (int i = 0; i < 2; ++i) {
            const int st = wave * 2 + i;
            const int tc = st * 32;                  // 32 grid columns per tile
            v8f acc = {};
#pragma unroll
            for (int k = 0; k < 4; ++k) {
                // A operand: stacked tile S (rows 0-7 = cols tc..tc+15,
                // rows 8-15 = cols tc+16..tc+31).  M = lane%16, K per half-wave.
                const int c0   = 4 * k + (hf ? 2 : 0);
                const int arow = 1 + (li & 7);
                const int acol = tc + (li >> 3) * 16 + c0;
                const v2f a = *(const v2f*)&h1[arow * NXP + acol];   // ds_load_b64
                acc = __builtin_amdgcn_wmma_f32_16x16x4_f32(
                        false, a, false, Bc[k], (short)0, acc, false, false);
            }
            // C/D layout: vgpr v, M = v + 8*(lane>=16) -> strip row v,
            // column block selected by half-wave.
            const int j = tc + hf * 16 + li;
#pragma unroll
            for (int v = 0; v < 8; ++v) {
                const float u  = h1[(v + 1) * NXP + j];
                const float up = h1[(v    ) * NXP + j];   // halo row when v==0
                const float dn = h1[(v + 2) * NXP + j];   // halo row when v==7
                float p = acc[v] + up + dn - 2.f * u;     // + vertical diff
                if (li == 0)  p += (j > 0)       ? h1[(v + 1) * NXP + j - 1] : 0.f;
                if (li == 15) p += (j < NXv - 1) ? h1[(v + 1) * NXP + j + 1] : 0.f;
                const float hn = 2.f * u - h2[(v + 1) * NXP + j] + cf_r[i][v] * p;
                h2[(v + 1) * NXP + j] = hn;
            }
        }
        __syncthreads();

        // ---- source injection (single thread: deterministic) -----------------
        if (tid == 0) {
#pragma unroll
            for (int s = 0; s < NSRCv; ++s) {
                const int sz = src_z[s];
                if (sz >= z0 && sz < z0 + ROWS)
                    h2[(sz - z0 + 1) * NXP + src_x[s]] += x[s * NTv + t];
            }
        }
        __syncthreads();

        // ---- receiver recording ----------------------------------------------
        for (int r = tid; r < NRECv; r += NTHR) {
            const int rz = rec_z[r];
            if (rz >= z0 && rz < z0 + ROWS)
                out[t * NRECv + r] = h2[(rz - z0 + 1) * NXP + rec_x[r]];
        }

        // ---- export my boundary rows of h_new (double-buffered by parity) ----
        float* gh = ghalo + (size_t)(t & 1) * (NBLK * 2 * NXv);
        {
            const int lr = side ? ROWS : 1;                    // LDS rows 1 / 8
            const f4v b = *(const f4v*)&h2[lr * NXP + seg * 4];
            *(f4v*)&gh[(wg * 2 + side) * NXv + seg * 4] = b;
        }
        __threadfence();
        grid_barrier(cnt, gen);

        // ---- import neighbor halos into h_new's halo rows (async -> LDS) -----
        if (side == 0) {
            if (wg > 0) {
                const float* gsrc = &gh[((wg - 1) * 2 + 1) * NXv + seg * 4];
                float*       ldst = &h2[0 * NXP + seg * 4];
#if USE_ASYNC
                __builtin_amdgcn_global_load_async_to_lds_b128(
                    (g_v4i_p)gsrc, (l_v4i_p)ldst, 0, 0);
#else
                *(f4v*)ldst = *(const f4v*)gsrc;
#endif
            }
        } else {
            if (wg < NBLK - 1) {
                const float* gsrc = &gh[((wg + 1) * 2 + 0) * NXv + seg * 4];
                float*       ldst = &h2[(ROWS + 1) * NXP + seg * 4];
#if USE_ASYNC
                __builtin_amdgcn_global_load_async_to_lds_b128(
                    (g_v4i_p)gsrc, (l_v4i_p)ldst, 0, 0);
#else
                *(f4v*)ldst = *(const f4v*)gsrc;
#endif
            }
        }
#if USE_ASYNC
        __builtin_amdgcn_s_wait_asynccnt(0);
#endif
        __syncthreads();

        float* tmp = h1; h1 = h2; h2 = tmp;   // (h_new, h1) -> (h1, h2)
    }
}

extern "C" void kernel_launch(void* const* d_in, const int* in_sizes, int n_in,
                              void* d_out, int out_size, void* d_ws, size_t ws_size,
                              hipStream_t stream) {
    const float* x     = (const float*)d_in[0];
    const float* vel   = (const float*)d_in[1];
    const int*   src_z = (const int*)d_in[2];
    const int*   src_x = (const int*)d_in[3];
    const int*   rec_z = (const int*)d_in[4];
    const int*   rec_x = (const int*)d_in[5];
    float* out = (float*)d_out;

    unsigned* bar   = (unsigned*)d_ws;
    float*    ghalo = (float*)((char*)d_ws + 256);   // 2 x 64 x 2 x 512 floats

    fdtd_init_ws<<<1, 32, 0, stream>>>(bar);
    fdtd_wave_persist<<<NBLK, NTHR, 0, stream>>>(
        x, vel, src_z, src_x, rec_z, rec_x, out, bar, ghalo);
}